// ArcFaceLoss_87986700026371
// MI455X (gfx1250) — compile-verified
//
#include <hip/hip_runtime.h>
#include <hip/hip_bf16.h>

// ---------------------------------------------------------------------------
// ArcFace loss, MI455X (gfx1250).
// Bandwidth-bound streaming reduction: 164 MB read @ 23.3 TB/s => ~7us floor.
// Direct coalesced float4 global loads (no LDS staging: zero reuse),
// WGP-scope prefetch one stride ahead (clamped address, branch-free),
// hardware-exp VALU, WMMA f32 16x16x4 as the cross-lane reducer.
// ---------------------------------------------------------------------------

#define NCLS   10000
#define NV4    (NCLS / 4)          // 2500 float4 per row
#define TPB    256
#define NFULL  ((NV4 / TPB) * TPB) // 2304: fully-active iterations
#define S_VAL  32.0f
#define M_VAL  0.5f
#define EPS_V  1e-7f

typedef __attribute__((ext_vector_type(2))) float v2f;
typedef __attribute__((ext_vector_type(8))) float v8f;

// Sum of all 32 lanes' v; every lane returns the total.
// V_WMMA_F32_16X16X4_F32 with B = ones:
//   A[m,0]=acc_m (lanes 0-15), A[m,2]=acc_{m+16} (lanes 16-31), rest 0
//   => D[m,n] = acc_m + acc_{m+16} for every n.
// Per-lane sum of the 8 D vgprs gives sum over M=0..7 (lanes 0-15) or
// M=8..15 (lanes 16-31); one shfl_xor(16) completes the reduction.
// Requires EXEC == all ones (call only from fully converged code).
__device__ __forceinline__ float wave_sum_f32(float v) {
#if __has_builtin(__builtin_amdgcn_wmma_f32_16x16x4_f32)
    v2f a; a[0] = v;    a[1] = 0.0f;
    v2f b; b[0] = 1.0f; b[1] = 1.0f;
    v8f c = {};
    c = __builtin_amdgcn_wmma_f32_16x16x4_f32(
        /*neg_a=*/false, a, /*neg_b=*/false, b,
        /*c_mod=*/(short)0, c, /*reuse_a=*/false, /*reuse_b=*/false);
    float s = ((c[0] + c[1]) + (c[2] + c[3])) + ((c[4] + c[5]) + (c[6] + c[7]));
    s += __shfl_xor(s, 16, 32);
    return s;
#else
    for (int off = 16; off > 0; off >>= 1) v += __shfl_xor(v, off, 32);
    return v;
#endif
}

// One 256-thread block (8 wave32) per row. Streams the row with float4
// loads (4 KB contiguous per block per iteration), accumulates exp(S*x),
// then computes the per-row ArcFace term into ws[row].
__global__ __launch_bounds__(TPB) void arcface_row_kernel(
    const float* __restrict__ x,     // [rows, NCLS]
    const int*   __restrict__ label, // [rows]
    float*       __restrict__ ws)    // [rows] per-row L_i
{
    const int row = blockIdx.x;
    const int tid = threadIdx.x;
    const float4* __restrict__ rp =
        reinterpret_cast<const float4*>(x + (size_t)row * NCLS);

    float acc = 0.0f;

    // Main loop: 9 iterations, all 256 threads active, no guards.
    for (int j = tid; j < NFULL; j += TPB) {
        // Branch-free clamped WGP-scope prefetch (address always in-bounds,
        // so the non-speculative WGP prefetch cannot fault).
        int jp = j + TPB; jp = (jp < NV4) ? jp : (NV4 - 1);
        __builtin_prefetch(&rp[jp], 0, 3);
        float4 v = rp[j];
        acc += __expf(S_VAL * v.x);
        acc += __expf(S_VAL * v.y);
        acc += __expf(S_VAL * v.z);
        acc += __expf(S_VAL * v.w);
    }
    // Tail: 196 of 256 threads active for the last partial chunk.
    {
        const int j = NFULL + tid;
        if (j < NV4) {
            float4 v = rp[j];
            acc += __expf(S_VAL * v.x);
            acc += __expf(S_VAL * v.y);
            acc += __expf(S_VAL * v.z);
            acc += __expf(S_VAL * v.w);
        }
    }

    // Fully reconverged here: EXEC all ones -> WMMA reduction is legal.
    float wsum = wave_sum_f32(acc);

    __shared__ float warp_part[TPB / 32];
    const int lane = tid & 31;
    const int wid  = tid >> 5;
    if (lane == 0) warp_part[wid] = wsum;
    __syncthreads();

    if (tid == 0) {
        float total = 0.0f;
        #pragma unroll
        for (int w = 0; w < TPB / 32; ++w) total += warp_part[w];

        const int   lab = label[row];
        const float tgt = x[(size_t)row * NCLS + lab];
        const float tc  = fminf(fmaxf(tgt, -1.0f + EPS_V), 1.0f - EPS_V);
        const float num = S_VAL * cosf(acosf(tc) + M_VAL);
        // excl_sum = row_sum - exp(S*tgt); use the same fast-exp as the loop
        // so the label column cancels consistently.
        const float excl  = total - __expf(S_VAL * tgt);
        const float denom = __expf(num) + excl;
        ws[row] = num - logf(denom);
    }
}

// Single-block deterministic mean: out = -(1/rows) * sum_i L_i.
__global__ __launch_bounds__(TPB) void arcface_reduce_kernel(
    const float* __restrict__ ws, float* __restrict__ out, int rows)
{
    const int tid = threadIdx.x;
    float acc = 0.0f;
    for (int j = tid; j < rows; j += TPB) acc += ws[j];

    float wsum = wave_sum_f32(acc);

    __shared__ float warp_part[TPB / 32];
    if ((tid & 31) == 0) warp_part[tid >> 5] = wsum;
    __syncthreads();

    if (tid == 0) {
        float total = 0.0f;
        #pragma unroll
        for (int w = 0; w < TPB / 32; ++w) total += warp_part[w];
        out[0] = -total / (float)rows;   // LOSS_WEIGHT == 1.0
    }
}

extern "C" void kernel_launch(void* const* d_in, const int* in_sizes, int n_in,
                              void* d_out, int out_size, void* d_ws, size_t ws_size,
                              hipStream_t stream) {
    const float* x     = (const float*)d_in[0];   // [rows, 10000] fp32
    const int*   label = (const int*)d_in[1];     // [rows] int
    float*       out   = (float*)d_out;
    float*       ws    = (float*)d_ws;            // rows * 4 bytes scratch

    const int rows = in_sizes[1];                 // 4096

    arcface_row_kernel<<<rows, TPB, 0, stream>>>(x, label, ws);
    arcface_reduce_kernel<<<1, TPB, 0, stream>>>(ws, out, rows);
}